// verletNetworks_46145128628937
// MI455X (gfx1250) — compile-verified
//
#include <hip/hip_runtime.h>

#define NN 50000
#define NE 1600000
#define NNP 50016                 // nodes padded to multiple of 32
#define NTILES_N (NNP / 32)       // 1563
#define HSTEP 0.1f
#define LDS_STRIDE 68             // 64 + 4 pad: conflict-free b64 frag reads

typedef __attribute__((ext_vector_type(2))) float v2f;
typedef __attribute__((ext_vector_type(8))) float v8f;

// ---------------------------------------------------------------------------
// WMMA fragment helpers.  Data in LDS is stored [row][k] with stride 68.
// Per ISA (32-bit A 16x4 / B 4x16): lane l holds (K = k0 + 2*(l>=16), K+1)
// for row m0+(l&15) (A) or column n=(l&15) (B) -> identical addressing.
// ---------------------------------------------------------------------------
__device__ __forceinline__ v2f ldfrag(const float* p, int row, int k0, int lane) {
  int kk = k0 + ((lane >> 4) << 1);
  return *(const v2f*)(p + row * LDS_STRIDE + kk);
}

// D = W[ m0..m0+15 ][0..63] * X[0..63][ n0..n0+15 ]   (K = 64 -> 16 wmma steps)
__device__ __forceinline__ v8f wmma_tile(const float* wlds, int m0,
                                         const float* xsub, int lane) {
  v8f acc = {0.f, 0.f, 0.f, 0.f, 0.f, 0.f, 0.f, 0.f};
  int m = m0 + (lane & 15);
  int n = lane & 15;
#pragma unroll
  for (int k0 = 0; k0 < 64; k0 += 4) {
    v2f a = ldfrag(wlds, m, k0, lane);
    v2f b = ldfrag(xsub, n, k0, lane);
    acc = __builtin_amdgcn_wmma_f32_16x16x4_f32(false, a, false, b,
                                                (short)0, acc, false, false);
  }
  return acc;
}

__device__ __forceinline__ void load_w(float* dst, const float* __restrict__ src,
                                       int rows, int tid, int nth) {
  for (int idx = tid; idx < rows * 64; idx += nth)
    dst[(idx >> 6) * LDS_STRIDE + (idx & 63)] = src[idx];
}

// Stage Ai = [grad ; ave] for 32 edges: lane l -> edge e0+l, LDS row l.
__device__ __forceinline__ void stage_edges(float* xt, const float* __restrict__ xnt,
                                            const int* __restrict__ iInd,
                                            const int* __restrict__ jInd,
                                            int e0, int lane) {
  int e = e0 + lane;
  int i = iInd[e], j = jInd[e];
  const float4* pi = (const float4*)(xnt + (size_t)i * 32);
  const float4* pj = (const float4*)(xnt + (size_t)j * 32);
  float* row = xt + lane * LDS_STRIDE;
#pragma unroll
  for (int q = 0; q < 8; ++q) {
    float4 a = pi[q], b = pj[q];
    float4 g, m;
    g.x = a.x - b.x; g.y = a.y - b.y; g.z = a.z - b.z; g.w = a.w - b.w;
    m.x = 0.5f * (a.x + b.x); m.y = 0.5f * (a.y + b.y);
    m.z = 0.5f * (a.z + b.z); m.w = 0.5f * (a.w + b.w);
    *(float4*)(row + q * 4) = g;
    *(float4*)(row + 32 + q * 4) = m;
  }
}

// Stage Bi columns v0..v0+31: lane l -> column v0+l, LDS row l.
__device__ __forceinline__ void stage_nodes(float* xt, const float* __restrict__ Bi,
                                            int v0, int lane) {
  int v = v0 + lane;
  float* row = xt + lane * LDS_STRIDE;
#pragma unroll 8
  for (int c = 0; c < 64; ++c) row[c] = Bi[(size_t)c * NNP + v];
}

// ---------------------------------------------------------------------------
// Pass 1 kernels: compute sum / sum^2 of t = K1 @ X  (deterministic hierarchy)
// ---------------------------------------------------------------------------
__global__ __launch_bounds__(128) void k_edge_stats(
    const float* __restrict__ W1, const float* __restrict__ xnt,
    const int* __restrict__ iInd, const int* __restrict__ jInd,
    float* __restrict__ part) {
  __shared__ float w1[64 * LDS_STRIDE];
  __shared__ float xt_s[4][32 * LDS_STRIDE];
  __shared__ float red1[4], red2[4];
  load_w(w1, W1, 64, threadIdx.x, 128);
  __syncthreads();
  int wv = threadIdx.x >> 5, lane = threadIdx.x & 31;
  int tile = blockIdx.x * 4 + wv;           // < 50000 by construction
  float* xt = xt_s[wv];
  stage_edges(xt, xnt, iInd, jInd, tile * 32, lane);
  float s1 = 0.f, s2 = 0.f;
#pragma unroll
  for (int ns = 0; ns < 2; ++ns) {
    const float* xb = xt + ns * 16 * LDS_STRIDE;
#pragma unroll
    for (int ms = 0; ms < 4; ++ms) {
      v8f acc = wmma_tile(w1, ms * 16, xb, lane);
#pragma unroll
      for (int r = 0; r < 8; ++r) { float t = acc[r]; s1 += t; s2 += t * t; }
    }
  }
#pragma unroll
  for (int off = 16; off > 0; off >>= 1) {
    s1 += __shfl_down(s1, off, 32);
    s2 += __shfl_down(s2, off, 32);
  }
  if (lane == 0) { red1[wv] = s1; red2[wv] = s2; }
  __syncthreads();
  if (threadIdx.x == 0) {
    part[2 * blockIdx.x]     = red1[0] + red1[1] + red1[2] + red1[3];
    part[2 * blockIdx.x + 1] = red2[0] + red2[1] + red2[2] + red2[3];
  }
}

__global__ __launch_bounds__(128) void k_node_stats(
    const float* __restrict__ W1, const float* __restrict__ Bi,
    float* __restrict__ part) {
  __shared__ float w1[64 * LDS_STRIDE];
  __shared__ float xt_s[4][32 * LDS_STRIDE];
  __shared__ float red1[4], red2[4];
  load_w(w1, W1, 64, threadIdx.x, 128);
  __syncthreads();
  int wv = threadIdx.x >> 5, lane = threadIdx.x & 31;
  int tile = blockIdx.x * 4 + wv;
  float s1 = 0.f, s2 = 0.f;
  if (tile < NTILES_N) {
    float* xt = xt_s[wv];
    stage_nodes(xt, Bi, tile * 32, lane);
#pragma unroll
    for (int ns = 0; ns < 2; ++ns) {
      const float* xb = xt + ns * 16 * LDS_STRIDE;
#pragma unroll
      for (int ms = 0; ms < 4; ++ms) {
        v8f acc = wmma_tile(w1, ms * 16, xb, lane);
#pragma unroll
        for (int r = 0; r < 8; ++r) { float t = acc[r]; s1 += t; s2 += t * t; }
      }
    }
  }
#pragma unroll
  for (int off = 16; off > 0; off >>= 1) {
    s1 += __shfl_down(s1, off, 32);
    s2 += __shfl_down(s2, off, 32);
  }
  if (lane == 0) { red1[wv] = s1; red2[wv] = s2; }
  __syncthreads();
  if (threadIdx.x == 0) {
    part[2 * blockIdx.x]     = red1[0] + red1[1] + red1[2] + red1[3];
    part[2 * blockIdx.x + 1] = red2[0] + red2[1] + red2[2] + red2[3];
  }
}

__global__ __launch_bounds__(256) void k_finalize(const float* __restrict__ part,
                                                  int g, float invcnt,
                                                  float* __restrict__ stat) {
  __shared__ float r1[256], r2[256];
  float s1 = 0.f, s2 = 0.f;
  for (int i = threadIdx.x; i < g; i += 256) { s1 += part[2 * i]; s2 += part[2 * i + 1]; }
  r1[threadIdx.x] = s1; r2[threadIdx.x] = s2;
  __syncthreads();
  for (int off = 128; off > 0; off >>= 1) {
    if (threadIdx.x < off) {
      r1[threadIdx.x] += r1[threadIdx.x + off];
      r2[threadIdx.x] += r2[threadIdx.x + off];
    }
    __syncthreads();
  }
  if (threadIdx.x == 0) {
    float mean = r1[0] * invcnt;
    float var = r2[0] * invcnt - mean * mean;
    stat[0] = mean;
    stat[1] = rsqrtf(var + 1e-5f);
  }
}

// ---------------------------------------------------------------------------
// Pass 2 kernels: recompute t, normalize+relu, second GEMM, residual update.
// ---------------------------------------------------------------------------
__global__ __launch_bounds__(64) void k_edge_apply(
    const float* __restrict__ W1, const float* __restrict__ W2,
    const float* __restrict__ xnt, const int* __restrict__ iInd,
    const int* __restrict__ jInd, float* __restrict__ xe,
    const float* __restrict__ stat) {
  __shared__ float w1[64 * LDS_STRIDE];
  __shared__ float w2[32 * LDS_STRIDE];
  __shared__ float xt_s[2][32 * LDS_STRIDE];
  __shared__ float ut_s[2][32 * LDS_STRIDE];
  load_w(w1, W1, 64, threadIdx.x, 64);
  load_w(w2, W2, 32, threadIdx.x, 64);
  __syncthreads();
  float mean = stat[0], rstd = stat[1];
  int wv = threadIdx.x >> 5, lane = threadIdx.x & 31;
  int tile = blockIdx.x * 2 + wv;           // exact: 25000 blocks * 2 = 50000
  int e0 = tile * 32;
  float* xt = xt_s[wv];
  float* ut = ut_s[wv];
  stage_edges(xt, xnt, iInd, jInd, e0, lane);
  int n = lane & 15;
  int hi8 = (lane >> 4) << 3;
#pragma unroll
  for (int ns = 0; ns < 2; ++ns) {
    const float* xb = xt + ns * 16 * LDS_STRIDE;
    float* ub = ut + ns * 16 * LDS_STRIDE;
#pragma unroll
    for (int ms = 0; ms < 4; ++ms) {
      v8f acc = wmma_tile(w1, ms * 16, xb, lane);
      float* urow = ub + n * LDS_STRIDE + ms * 16 + hi8;
#pragma unroll
      for (int r = 0; r < 8; ++r) {
        float u = (acc[r] - mean) * rstd;
        urow[r] = u > 0.f ? u : 0.f;
      }
    }
    int e = e0 + ns * 16 + n;
#pragma unroll
    for (int ms = 0; ms < 2; ++ms) {
      v8f acc = wmma_tile(w2, ms * 16, ub, lane);
#pragma unroll
      for (int r = 0; r < 8; ++r) {
        size_t off = (size_t)(ms * 16 + hi8 + r) * NE + e;
        xe[off] += HSTEP * acc[r];
      }
    }
  }
}

__global__ __launch_bounds__(64) void k_node_apply(
    const float* __restrict__ W1, const float* __restrict__ W2,
    const float* __restrict__ Bi, float* __restrict__ xn,
    float* __restrict__ xnt, const float* __restrict__ stat) {
  __shared__ float w1[64 * LDS_STRIDE];
  __shared__ float w2[32 * LDS_STRIDE];
  __shared__ float xt_s[2][32 * LDS_STRIDE];
  __shared__ float ut_s[2][32 * LDS_STRIDE];
  load_w(w1, W1, 64, threadIdx.x, 64);
  load_w(w2, W2, 32, threadIdx.x, 64);
  __syncthreads();
  int wv = threadIdx.x >> 5, lane = threadIdx.x & 31;
  int tile = blockIdx.x * 2 + wv;
  if (tile >= NTILES_N) return;
  float mean = stat[0], rstd = stat[1];
  int v0 = tile * 32;
  float* xt = xt_s[wv];
  float* ut = ut_s[wv];
  stage_nodes(xt, Bi, v0, lane);
  int n = lane & 15;
  int hi8 = (lane >> 4) << 3;
#pragma unroll
  for (int ns = 0; ns < 2; ++ns) {
    const float* xb = xt + ns * 16 * LDS_STRIDE;
    float* ub = ut + ns * 16 * LDS_STRIDE;
#pragma unroll
    for (int ms = 0; ms < 4; ++ms) {
      v8f acc = wmma_tile(w1, ms * 16, xb, lane);
      float* urow = ub + n * LDS_STRIDE + ms * 16 + hi8;
#pragma unroll
      for (int r = 0; r < 8; ++r) {
        float u = (acc[r] - mean) * rstd;
        urow[r] = u > 0.f ? u : 0.f;
      }
    }
    int v = v0 + ns * 16 + n;
#pragma unroll
    for (int ms = 0; ms < 2; ++ms) {
      v8f acc = wmma_tile(w2, ms * 16, ub, lane);
      if (v < NN) {
#pragma unroll
        for (int r = 0; r < 8; ++r) {
          int ch = ms * 16 + hi8 + r;
          float nv = xn[(size_t)ch * NN + v] + HSTEP * acc[r];
          xn[(size_t)ch * NN + v] = nv;
          xnt[(size_t)v * 32 + ch] = nv;
        }
      }
    }
  }
}

// ---------------------------------------------------------------------------
// Open / close / scatter / zero (memory-bound VALU kernels)
// ---------------------------------------------------------------------------
__global__ __launch_bounds__(256) void k_open_node(
    const float* __restrict__ xin, const float* __restrict__ W,
    float* __restrict__ xn, float* __restrict__ xnt) {
  __shared__ float w[32 * 16];
  for (int i = threadIdx.x; i < 512; i += 256) w[i] = W[i];
  __syncthreads();
  int v = blockIdx.x * 256 + threadIdx.x;
  if (v >= NN) return;
  float in[16];
#pragma unroll
  for (int k = 0; k < 16; ++k) in[k] = xin[(size_t)k * NN + v];
  float out[32];
#pragma unroll
  for (int c = 0; c < 32; ++c) {
    float a = 0.f;
#pragma unroll
    for (int k = 0; k < 16; ++k) a = fmaf(w[c * 16 + k], in[k], a);
    out[c] = a;
    xn[(size_t)c * NN + v] = a;
  }
#pragma unroll
  for (int q = 0; q < 8; ++q) {
    float4 o; o.x = out[4 * q]; o.y = out[4 * q + 1]; o.z = out[4 * q + 2]; o.w = out[4 * q + 3];
    *(float4*)(xnt + (size_t)v * 32 + q * 4) = o;
  }
}

__global__ __launch_bounds__(256) void k_open_edge(
    const float* __restrict__ xin, const float* __restrict__ W,
    float* __restrict__ xe) {
  __shared__ float w[32 * 16];
  for (int i = threadIdx.x; i < 512; i += 256) w[i] = W[i];
  __syncthreads();
  int e = blockIdx.x * 256 + threadIdx.x;
  if (e >= NE) return;
  float in[16];
#pragma unroll
  for (int k = 0; k < 16; ++k) in[k] = xin[(size_t)k * NE + e];
#pragma unroll
  for (int c = 0; c < 32; ++c) {
    float a = 0.f;
#pragma unroll
    for (int k = 0; k < 16; ++k) a = fmaf(w[c * 16 + k], in[k], a);
    xe[(size_t)c * NE + e] = a;
  }
}

__global__ __launch_bounds__(256) void k_close(
    const float* __restrict__ x, const float* __restrict__ W,
    float* __restrict__ out, int ncols) {
  __shared__ float w[32 * 32];
  for (int i = threadIdx.x; i < 1024; i += 256) w[i] = W[i];
  __syncthreads();
  int v = blockIdx.x * 256 + threadIdx.x;
  if (v >= ncols) return;
  float in[32];
#pragma unroll
  for (int c = 0; c < 32; ++c) in[c] = x[(size_t)c * ncols + v];
#pragma unroll
  for (int o = 0; o < 32; ++o) {
    float a = 0.f;
#pragma unroll
    for (int c = 0; c < 32; ++c) a = fmaf(w[o * 32 + c], in[c], a);
    out[(size_t)o * ncols + v] = a;
  }
}

__global__ __launch_bounds__(256) void k_zero4(float* __restrict__ p, int n4) {
  int i = blockIdx.x * 256 + threadIdx.x;
  if (i < n4) {
    float4 z; z.x = z.y = z.z = z.w = 0.f;
    *(float4*)(p + (size_t)i * 4) = z;
  }
}

// Bi = [edge_div ; edge_ave] scatter (grid.y = channel)
__global__ __launch_bounds__(256) void k_scatter(
    const float* __restrict__ xe, const int* __restrict__ iInd,
    const int* __restrict__ jInd, float* __restrict__ Bi) {
  int c = blockIdx.y;
  int e = blockIdx.x * 256 + threadIdx.x;
  if (e >= NE) return;
  float v = xe[(size_t)c * NE + e];
  int i = iInd[e], j = jInd[e];
  float h = 0.5f * v;
  atomicAdd(&Bi[(size_t)c * NNP + i], v);
  atomicAdd(&Bi[(size_t)c * NNP + j], -v);
  atomicAdd(&Bi[(size_t)(32 + c) * NNP + i], h);
  atomicAdd(&Bi[(size_t)(32 + c) * NNP + j], h);
}

// ---------------------------------------------------------------------------
extern "C" void kernel_launch(void* const* d_in, const int* in_sizes, int n_in,
                              void* d_out, int out_size, void* d_ws, size_t ws_size,
                              hipStream_t stream) {
  const float* xn_in   = (const float*)d_in[0];
  const float* xe_in   = (const float*)d_in[1];
  const int*   iInd    = (const int*)d_in[2];
  const int*   jInd    = (const int*)d_in[3];
  const float* KNopen  = (const float*)d_in[4];
  const float* KEopen  = (const float*)d_in[5];
  const float* KE1     = (const float*)d_in[6];
  const float* KE2     = (const float*)d_in[7];
  const float* KN1     = (const float*)d_in[8];
  const float* KN2     = (const float*)d_in[9];
  const float* KNclose = (const float*)d_in[10];
  const float* KEclose = (const float*)d_in[11];

  float* ws = (float*)d_ws;
  float* XE    = ws;                                 // 32 * NE  = 51,200,000
  float* XN    = XE + (size_t)32 * NE;               // 32 * NN  =  1,600,000
  float* XNT   = XN + (size_t)32 * NN;               // NN * 32  =  1,600,000
  float* BI    = XNT + (size_t)32 * NN;              // 64 * NNP =  3,201,024
  float* EPART = BI + (size_t)64 * NNP;              // 2 * 12500
  float* NPART = EPART + 25008;                      // 2 * 391
  float* ESTAT = NPART + 800;                        // 2
  float* NSTAT = ESTAT + 8;                          // 2

  float* out_xn = (float*)d_out;                     // 32 * NN
  float* out_xe = out_xn + (size_t)32 * NN;          // 32 * NE

  k_open_node<<<(NN + 255) / 256, 256, 0, stream>>>(xn_in, KNopen, XN, XNT);
  k_open_edge<<<NE / 256, 256, 0, stream>>>(xe_in, KEopen, XE);

  const float inv_e = (float)(1.0 / (64.0 * (double)NE));
  const float inv_n = (float)(1.0 / (64.0 * (double)NN));
  const int GE_STATS = NE / (32 * 4);                // 12500
  const int GN_STATS = (NTILES_N + 3) / 4;           // 391
  const int GE_APPLY = NE / (32 * 2);                // 25000
  const int GN_APPLY = (NTILES_N + 1) / 2;           // 782

  for (int L = 0; L < 4; ++L) {
    const float* ke1 = KE1 + (size_t)L * 64 * 64;
    const float* ke2 = KE2 + (size_t)L * 32 * 64;
    const float* kn1 = KN1 + (size_t)L * 64 * 64;
    const float* kn2 = KN2 + (size_t)L * 32 * 64;

    k_edge_stats<<<GE_STATS, 128, 0, stream>>>(ke1, XNT, iInd, jInd, EPART);
    k_finalize<<<1, 256, 0, stream>>>(EPART, GE_STATS, inv_e, ESTAT);
    k_edge_apply<<<GE_APPLY, 64, 0, stream>>>(ke1, ke2, XNT, iInd, jInd, XE, ESTAT);

    k_zero4<<<(64 * NNP / 4 + 255) / 256, 256, 0, stream>>>(BI, 64 * NNP / 4);
    dim3 sg(NE / 256, 32);
    k_scatter<<<sg, 256, 0, stream>>>(XE, iInd, jInd, BI);

    k_node_stats<<<GN_STATS, 128, 0, stream>>>(kn1, BI, NPART);
    k_finalize<<<1, 256, 0, stream>>>(NPART, GN_STATS, inv_n, NSTAT);
    k_node_apply<<<GN_APPLY, 64, 0, stream>>>(kn1, kn2, BI, XN, XNT, NSTAT);
  }

  k_close<<<(NN + 255) / 256, 256, 0, stream>>>(XN, KNclose, out_xn, NN);
  k_close<<<NE / 256, 256, 0, stream>>>(XE, KEclose, out_xe, NE);
}